// SimilarityBlock_40484361732250
// MI455X (gfx1250) — compile-verified
//
#include <hip/hip_runtime.h>
#include <hip/hip_bf16.h>
#include <math.h>

// ---------------------------------------------------------------------------
// GNN stack for MI455X (gfx1250, wave32).
//   - fp32 WMMA 16x16x4 for all dense GEMMs (D=64 -> 16 k-steps)
//   - one wave = one 16-row x 64-col output tile, weights staged in LDS
//   - edge gather/scatter: 16 threads/edge, float4 gathers, f32 global atomics
//   - divides replaced by v_rcp_f32 (softmax-tolerant 1-ulp reciprocal)
// Workspace layout (floats): hA[N*64] hB[N*64] h2[N*64] deg[N] as[N] ad[N]
//                            m[N] z[N] ew[E+N] (~93 MB for N=100k)
// ---------------------------------------------------------------------------

typedef __attribute__((ext_vector_type(2))) float v2f;
typedef __attribute__((ext_vector_type(8))) float v8f;

static __device__ inline v8f wmma4(v2f a, v2f b, v8f c) {
  // D = A(16x4, f32) * B(4x16, f32) + C(16x16, f32)
  return __builtin_amdgcn_wmma_f32_16x16x4_f32(false, a, false, b, (short)0, c,
                                               false, false);
}

static __device__ inline float fast_rcp(float x) {
  return __builtin_amdgcn_rcpf(x);  // v_rcp_f32
}

static __device__ inline void atomicMaxF(float* addr, float val) {
  if (val >= 0.0f)
    atomicMax((int*)addr, __float_as_int(val));
  else
    atomicMin((unsigned int*)addr, (unsigned int)__float_as_int(val));
}

// ------------------------- SAGE fused GEMM ---------------------------------
// out = (msg/deg) @ Wl^T + bl + hin @ Wr^T   [+ ReLU]
__global__ void __launch_bounds__(256)
sage_gemm_wmma(const float* __restrict__ hin, const float* __restrict__ msg,
               const float* __restrict__ deg, const float* __restrict__ Wl,
               const float* __restrict__ bl, const float* __restrict__ Wr,
               float* __restrict__ out, int n, int act) {
  __shared__ float sWl[64 * 64];
  __shared__ float sWr[64 * 64];
  for (int i = threadIdx.x; i < 64 * 64; i += blockDim.x) {
    sWl[i] = Wl[i];
    sWr[i] = Wr[i];
  }
  __syncthreads();

  int wave = (int)((blockIdx.x * blockDim.x + threadIdx.x) >> 5);
  int lane = threadIdx.x & 31;
  int tiles = (n + 15) >> 4;
  if (wave >= tiles) return;  // wave-uniform branch: EXEC stays all-ones

  int row0 = wave << 4;
  int lrow = lane & 15;
  int khalf = lane >> 4;  // A/B fragment K-pair select
  int r = row0 + lrow;
  int rr = r < n ? r : n - 1;

  float scale = fast_rcp(fmaxf(deg[rr], 1.0f));
  const float* mrow = msg + (size_t)rr * 64;
  const float* hrow = hin + (size_t)rr * 64;

  v8f acc[4] = {};
#pragma unroll
  for (int ks = 0; ks < 16; ++ks) {
    int kb = ks * 4 + khalf * 2;
    v2f a1, a2;
    a1.x = mrow[kb] * scale;
    a1.y = mrow[kb + 1] * scale;
    a2.x = hrow[kb];
    a2.y = hrow[kb + 1];
#pragma unroll
    for (int t = 0; t < 4; ++t) {
      int c = t * 16 + lrow;
      v2f b1, b2;
      b1.x = sWl[c * 64 + kb];
      b1.y = sWl[c * 64 + kb + 1];
      b2.x = sWr[c * 64 + kb];
      b2.y = sWr[c * 64 + kb + 1];
      acc[t] = wmma4(a1, b1, acc[t]);
      acc[t] = wmma4(a2, b2, acc[t]);
    }
  }
#pragma unroll
  for (int t = 0; t < 4; ++t) {
    int col = t * 16 + lrow;
    float bv = bl[col];
#pragma unroll
    for (int q = 0; q < 8; ++q) {
      int orow = row0 + q + khalf * 8;
      if (orow < n) {
        float v = acc[t][q] + bv;
        if (act) v = fmaxf(v, 0.0f);
        out[(size_t)orow * 64 + col] = v;
      }
    }
  }
}

// ------------------------- generic GEMM ------------------------------------
// out[n,O] = act(in[n,K] @ W[O,K]^T + b)   (K,O multiples of 4/16, K,O<=64)
__global__ void __launch_bounds__(256)
gemm_wmma(const float* __restrict__ in, const float* __restrict__ W,
          const float* __restrict__ bias, float* __restrict__ out, int n, int K,
          int O, int act) {
  __shared__ float sW[64 * 64];
  for (int i = threadIdx.x; i < O * K; i += blockDim.x) sW[i] = W[i];
  __syncthreads();

  int wave = (int)((blockIdx.x * blockDim.x + threadIdx.x) >> 5);
  int lane = threadIdx.x & 31;
  int tiles = (n + 15) >> 4;
  if (wave >= tiles) return;

  int row0 = wave << 4;
  int lrow = lane & 15;
  int khalf = lane >> 4;
  int r = row0 + lrow;
  int rr = r < n ? r : n - 1;
  const float* arow = in + (size_t)rr * K;

  int ksteps = K >> 2;
  int ntiles = O >> 4;
  v8f acc[4] = {};
  for (int ks = 0; ks < ksteps; ++ks) {
    int kb = ks * 4 + khalf * 2;
    v2f a;
    a.x = arow[kb];
    a.y = arow[kb + 1];
    for (int t = 0; t < ntiles; ++t) {
      int c = t * 16 + lrow;
      v2f b;
      b.x = sW[c * K + kb];
      b.y = sW[c * K + kb + 1];
      acc[t] = wmma4(a, b, acc[t]);
    }
  }
  for (int t = 0; t < ntiles; ++t) {
    int col = t * 16 + lrow;
    float bv = bias ? bias[col] : 0.0f;
#pragma unroll
    for (int q = 0; q < 8; ++q) {
      int orow = row0 + q + khalf * 8;
      if (orow < n) {
        float v = acc[t][q] + bv;
        if (act) v = fmaxf(v, 0.0f);
        out[(size_t)orow * O + col] = v;
      }
    }
  }
}

// ------------------------- graph kernels -----------------------------------
__global__ void sage_scatter(const float* __restrict__ h,
                             const int* __restrict__ src,
                             const int* __restrict__ dst,
                             float* __restrict__ msg, float* __restrict__ deg,
                             int E) {
  long long idx = (long long)blockIdx.x * blockDim.x + threadIdx.x;
  long long e = idx >> 4;
  if (e >= E) return;
  int f0 = (int)(idx & 15) * 4;
  int s = src[e], d = dst[e];
  float4 hv = *reinterpret_cast<const float4*>(h + (size_t)s * 64 + f0);
  float* mp = msg + (size_t)d * 64 + f0;
  atomicAdd(mp + 0, hv.x);
  atomicAdd(mp + 1, hv.y);
  atomicAdd(mp + 2, hv.z);
  atomicAdd(mp + 3, hv.w);
  if (f0 == 0) atomicAdd(deg + d, 1.0f);
}

__global__ void gat_alpha(const float* __restrict__ h2,
                          const float* __restrict__ avs,
                          const float* __restrict__ avd, float* __restrict__ as_,
                          float* __restrict__ ad_, int n) {
  int i = blockIdx.x * blockDim.x + threadIdx.x;
  if (i >= n) return;
  const float* hr = h2 + (size_t)i * 64;
  float s = 0.f, d = 0.f;
#pragma unroll
  for (int f = 0; f < 64; ++f) {
    float hv = hr[f];
    s += hv * avs[f];
    d += hv * avd[f];
  }
  as_[i] = s;
  ad_[i] = d;
}

__global__ void gat_edge_max(const int* __restrict__ src,
                             const int* __restrict__ dst,
                             const float* __restrict__ as_,
                             const float* __restrict__ ad_,
                             float* __restrict__ ew, float* __restrict__ m,
                             int E, int n) {
  int i = blockIdx.x * blockDim.x + threadIdx.x;
  int EN = E + n;
  if (i >= EN) return;
  int s, d;
  if (i < E) {
    s = src[i];
    d = dst[i];
  } else {
    s = d = i - E;
  }
  float e = as_[s] + ad_[d];
  e = e > 0.0f ? e : 0.2f * e;  // leaky_relu(0.2)
  ew[i] = e;
  atomicMaxF(m + d, e);
}

__global__ void gat_edge_exp(const int* __restrict__ dst, float* __restrict__ ew,
                             const float* __restrict__ m, float* __restrict__ z,
                             int E, int n) {
  int i = blockIdx.x * blockDim.x + threadIdx.x;
  int EN = E + n;
  if (i >= EN) return;
  int d = (i < E) ? dst[i] : (i - E);
  float p = __expf(ew[i] - m[d]);
  ew[i] = p;
  atomicAdd(z + d, p);
}

// alpha_e = ew[e] * rcp(z[dst]) precomputed once per edge (not per chunk)
__global__ void gat_edge_alpha(const int* __restrict__ dst,
                               float* __restrict__ ew,
                               const float* __restrict__ z, int E, int n) {
  int i = blockIdx.x * blockDim.x + threadIdx.x;
  int EN = E + n;
  if (i >= EN) return;
  int d = (i < E) ? dst[i] : (i - E);
  ew[i] = ew[i] * fast_rcp(z[d]);  // v_rcp_f32, no div sequence
}

__global__ void gat_edge_scatter(const int* __restrict__ src,
                                 const int* __restrict__ dst,
                                 const float* __restrict__ alpha,
                                 const float* __restrict__ h2,
                                 float* __restrict__ out, int E, int n) {
  long long idx = (long long)blockIdx.x * blockDim.x + threadIdx.x;
  long long e = idx >> 4;
  long long EN = (long long)E + n;
  if (e >= EN) return;
  int f0 = (int)(idx & 15) * 4;
  int s, d;
  if (e < E) {
    s = src[e];
    d = dst[e];
  } else {
    s = d = (int)(e - E);
  }
  float w = alpha[e];
  float4 hv = *reinterpret_cast<const float4*>(h2 + (size_t)s * 64 + f0);
  float* op = out + (size_t)d * 64 + f0;
  atomicAdd(op + 0, w * hv.x);
  atomicAdd(op + 1, w * hv.y);
  atomicAdd(op + 2, w * hv.z);
  atomicAdd(op + 3, w * hv.w);
}

__global__ void bias_act(float* __restrict__ h, const float* __restrict__ b,
                         long long total, int D, int act) {
  long long i = (long long)blockIdx.x * blockDim.x + threadIdx.x;
  if (i >= total) return;
  int f = (int)(i % D);
  float v = h[i] + b[f];
  if (act) v = fmaxf(v, 0.0f);
  h[i] = v;
}

__global__ void fillf(float* __restrict__ p, float v, long long nelems) {
  long long i = (long long)blockIdx.x * blockDim.x + threadIdx.x;
  if (i < nelems) p[i] = v;
}

// ---------------------------------------------------------------------------
extern "C" void kernel_launch(void* const* d_in, const int* in_sizes, int n_in,
                              void* d_out, int out_size, void* d_ws,
                              size_t ws_size, hipStream_t stream) {
  const float* x = (const float*)d_in[0];
  const int* ei = (const int*)d_in[1];
  const float* sWl = (const float*)d_in[2];
  const float* sbl = (const float*)d_in[3];
  const float* sWr = (const float*)d_in[4];
  const float* gW = (const float*)d_in[5];
  const float* gas = (const float*)d_in[6];
  const float* gad = (const float*)d_in[7];
  const float* gb = (const float*)d_in[8];
  const float* W1 = (const float*)d_in[9];
  const float* b1 = (const float*)d_in[10];
  const float* W2 = (const float*)d_in[11];
  const float* b2 = (const float*)d_in[12];
  const float* W3 = (const float*)d_in[13];
  const float* b3 = (const float*)d_in[14];

  const int n = in_sizes[0] / 64;
  const int E = in_sizes[1] / 2;
  const int* src = ei;
  const int* dst = ei + E;

  float* ws = (float*)d_ws;
  size_t o = 0;
  float* hA = ws + o;  o += (size_t)n * 64;
  float* hB = ws + o;  o += (size_t)n * 64;
  float* h2 = ws + o;  o += (size_t)n * 64;  // msg buffer / GAT h' / MLP tmp
  float* deg = ws + o; o += n;
  float* as_ = ws + o; o += n;
  float* ad_ = ws + o; o += n;
  float* m = ws + o;   o += n;
  float* z = ws + o;   o += n;
  float* ew = ws + o;  o += (size_t)E + n;

  const int tiles = (n + 15) / 16;
  const dim3 gemmGrid((tiles + 7) / 8);  // 8 waves (one 16-row tile each)/block
  const int EN = E + n;
  const int eScatBlk = (int)(((long long)E * 16 + 255) / 256);
  const int eScatBlkSL = (int)(((long long)EN * 16 + 255) / 256);
  const int nBlk = (n + 255) / 256;
  const int eBlk = (EN + 255) / 256;
  const long long nd = (long long)n * 64;
  const int ndBlk = (int)((nd + 255) / 256);

  const float* cur = x;
  float* nxt = hA;

  // ---- GraphSAGE x4 ----
  for (int l = 0; l < 4; ++l) {
    hipMemsetAsync(h2, 0, (size_t)n * 64 * sizeof(float), stream);
    hipMemsetAsync(deg, 0, (size_t)n * sizeof(float), stream);
    sage_scatter<<<eScatBlk, 256, 0, stream>>>(cur, src, dst, h2, deg, E);
    sage_gemm_wmma<<<gemmGrid, 256, 0, stream>>>(
        cur, h2, deg, sWl + (size_t)l * 4096, sbl + (size_t)l * 64,
        sWr + (size_t)l * 4096, nxt, n, (l < 3) ? 1 : 0);
    cur = nxt;
    nxt = (cur == hA) ? hB : hA;
  }

  // ---- GAT x4 (with self-loops) ----
  for (int l = 0; l < 4; ++l) {
    gemm_wmma<<<gemmGrid, 256, 0, stream>>>(cur, gW + (size_t)l * 4096, nullptr,
                                            h2, n, 64, 64, 0);
    gat_alpha<<<nBlk, 256, 0, stream>>>(h2, gas + (size_t)l * 64,
                                        gad + (size_t)l * 64, as_, ad_, n);
    fillf<<<nBlk, 256, 0, stream>>>(m, -INFINITY, n);
    hipMemsetAsync(z, 0, (size_t)n * sizeof(float), stream);
    gat_edge_max<<<eBlk, 256, 0, stream>>>(src, dst, as_, ad_, ew, m, E, n);
    gat_edge_exp<<<eBlk, 256, 0, stream>>>(dst, ew, m, z, E, n);
    gat_edge_alpha<<<eBlk, 256, 0, stream>>>(dst, ew, z, E, n);
    hipMemsetAsync(nxt, 0, (size_t)n * 64 * sizeof(float), stream);
    gat_edge_scatter<<<eScatBlkSL, 256, 0, stream>>>(src, dst, ew, h2, nxt, E,
                                                     n);
    bias_act<<<ndBlk, 256, 0, stream>>>(nxt, gb + (size_t)l * 64, nd, 64,
                                        (l < 3) ? 1 : 0);
    cur = nxt;
    nxt = (cur == hA) ? hB : hA;
  }

  // ---- Projection MLP ----
  float* t1 = nxt;  // free ping-pong buffer
  gemm_wmma<<<gemmGrid, 256, 0, stream>>>(cur, W1, b1, t1, n, 64, 64, 1);
  gemm_wmma<<<gemmGrid, 256, 0, stream>>>(t1, W2, b2, h2, n, 64, 32, 1);
  gemm_wmma<<<gemmGrid, 256, 0, stream>>>(h2, W3, b3, (float*)d_out, n, 32, 16,
                                          0);
}